// TriangleScene_52999896433062
// MI455X (gfx1250) — compile-verified
//
#include <hip/hip_runtime.h>
#include <hip/hip_bf16.h>

// ---------------------------------------------------------------------------
// Order-2 image-method path validation for MI455X (gfx1250, wave32).
//  * TDM tensor_load_to_lds stages the vertex array into LDS (TENSORcnt).
//  * global_load_async_to_lds_b128 stages the triangle index buffer (ASYNCcnt).
//  * Per-block LDS precompute of (v0,e1,e2) per triangle, SoA layout.
//  * One wave per (tx,rx,candidate): lanes split the 400 triangles for the
//    Moller-Trumbore occlusion test, reduced with ballot_w32.
// ---------------------------------------------------------------------------

#define N_TRI      400
#define N_VERT     900
#define PATH_ORDER 2
#define MT_EPS     1e-6f
#define MT_HIT_T   0.999f

typedef unsigned int       u32;
typedef unsigned long long u64;
typedef __attribute__((ext_vector_type(4))) u32 u32x4;
typedef __attribute__((ext_vector_type(8))) u32 u32x8;
typedef __attribute__((address_space(3))) void lds_void;

// ---------------- small float3 helpers ----------------
struct V3 { float x, y, z; };
__device__ __forceinline__ V3 v3(float a, float b, float c) { V3 r; r.x = a; r.y = b; r.z = c; return r; }
__device__ __forceinline__ V3 vsub(V3 a, V3 b) { return v3(a.x - b.x, a.y - b.y, a.z - b.z); }
__device__ __forceinline__ V3 vadd(V3 a, V3 b) { return v3(a.x + b.x, a.y + b.y, a.z + b.z); }
__device__ __forceinline__ V3 vmul(V3 a, float s) { return v3(a.x * s, a.y * s, a.z * s); }
__device__ __forceinline__ float vdot(V3 a, V3 b) { return __builtin_fmaf(a.x, b.x, __builtin_fmaf(a.y, b.y, a.z * b.z)); }
__device__ __forceinline__ V3 vcross(V3 a, V3 b) {
    return v3(a.y * b.z - a.z * b.y, a.z * b.x - a.x * b.z, a.x * b.y - a.y * b.x);
}

// mirror point p about plane (mv, mn):  p - 2*dot(p-mv,mn)*mn
__device__ __forceinline__ V3 mirror_pt(V3 p, V3 mv, V3 mn) {
    float k = 2.0f * vdot(vsub(p, mv), mn);
    return vsub(p, vmul(mn, k));
}
// intersect segment (img -> cur) with plane (mv, mn): img + t*(cur-img)
__device__ __forceinline__ V3 plane_line(V3 cur, V3 img, V3 mv, V3 mn) {
    V3 u = vsub(cur, img);
    float t = vdot(vsub(mv, img), mn) / vdot(u, mn);
    return vadd(img, vmul(u, t));
}
// point-in-triangle via same-side normals test (matches reference)
__device__ __forceinline__ bool tri_contains(V3 a, V3 b, V3 c, V3 v) {
    V3 u0 = vsub(a, v), u1 = vsub(b, v), u2 = vsub(c, v);
    V3 c0 = vcross(u0, u1), c1 = vcross(u1, u2), c2 = vcross(u2, u0);
    return (vdot(c0, c1) >= 0.0f) & (vdot(c1, c2) >= 0.0f) & (vdot(c2, c0) >= 0.0f);
}

// ---------------- CDNA5 data movement helpers ----------------
__device__ __forceinline__ u32 lds_byte_off(const void* p) {
    return (u32)(u64)(lds_void*)p;   // generic -> addrspace(3) -> 32-bit LDS offset
}
__device__ __forceinline__ u32 sgprify(u32 x) {
    return (u32)__builtin_amdgcn_readfirstlane((int)x);
}

// Per-lane async global->LDS copy of 16B (ASYNCcnt), GVS addressing mode.
__device__ __forceinline__ void async_load_b128(void* lds_dst, const void* gbase, u32 gbyteoff) {
    asm volatile("global_load_async_to_lds_b128 %0, %1, %2"
                 :: "v"(lds_byte_off(lds_dst)), "v"(gbyteoff), "s"((u64)gbase)
                 : "memory");
}

// TDM: DMA a contiguous run of `ndw` dwords global->LDS as a (ndw x 1) 2D tile.
// D# group0/group1 built per CDNA5 ISA; 2-group (<=2D) form.
__device__ __forceinline__ void tdm_load_1d_dwords(void* lds_dst, const void* gsrc, u32 ndw) {
    u64 ga = (u64)gsrc;
    u32x4 g0;
    g0.x = sgprify(1u);                                            // count=1, user mode
    g0.y = sgprify(lds_byte_off(lds_dst));                         // lds_addr
    g0.z = sgprify((u32)ga);                                       // global_addr[31:0]
    g0.w = sgprify(((u32)(ga >> 32) & 0x01FFFFFFu) | 0x80000000u); // addr[56:32] | type=2
    u32x8 g1;
    g1.s0 = sgprify(0x00020000u);                 // data_size=2 (4B), no multicast/pad/iter
    g1.s1 = sgprify((ndw & 0xFFFFu) << 16);       // tensor_dim0[15:0]
    g1.s2 = sgprify(((ndw >> 16) & 0xFFFFu) | (1u << 16)); // tensor_dim0[31:16], tensor_dim1=1
    g1.s3 = sgprify((ndw & 0xFFFFu) << 16);       // tile_dim0 = ndw
    g1.s4 = sgprify(1u);                          // tile_dim1=1, tile_dim2=0
    g1.s5 = sgprify(ndw);                         // tensor_dim0_stride[31:0]
    g1.s6 = sgprify((ndw & 0xFFFFu) << 16);       // stride0[47:32]=0 | tensor_dim1_stride[15:0]
    g1.s7 = sgprify(ndw >> 16);                   // tensor_dim1_stride[47:16]
    asm volatile("tensor_load_to_lds %0, %1" :: "s"(g0), "s"(g1) : "memory");
}

// ---------------------------------------------------------------------------
__global__ __launch_bounds__(256) void rt_paths_kernel(
    const float* __restrict__ g_tx, const float* __restrict__ g_rx,
    const float* __restrict__ g_verts, const float* __restrict__ g_norms,
    const int* __restrict__ g_tris, const int* __restrict__ g_pc,
    float* __restrict__ out, int ntx, int nrx, int npc, int total)
{
    __shared__ __align__(16) float sh_verts[N_VERT * 3];   // 10800 B
    __shared__ __align__(16) int   sh_tris[N_TRI * 3];     //  4800 B
    __shared__ float sh_pre[9][N_TRI];                     // 14400 B SoA: v0.xyz,e1.xyz,e2.xyz

    // ---- Stage 0: scene -> LDS (TDM for vertices, async copies for indices) ----
    if (threadIdx.x < 32u) {                   // wave 0 issues the tensor DMA
        tdm_load_1d_dwords(sh_verts, g_verts, N_VERT * 3);
        __builtin_amdgcn_s_wait_tensorcnt(0);
    }
    for (u32 i = threadIdx.x; i < (N_TRI * 3) / 4; i += blockDim.x)
        async_load_b128(&sh_tris[i * 4], g_tris, i * 16u);
    asm volatile("s_wait_asynccnt 0" ::: "memory");
    __syncthreads();

    // ---- Stage 1: precompute v0,e1,e2 per triangle into SoA LDS ----
    for (u32 t = threadIdx.x; t < N_TRI; t += blockDim.x) {
        int i0 = sh_tris[3 * t + 0], i1 = sh_tris[3 * t + 1], i2 = sh_tris[3 * t + 2];
        float ax = sh_verts[3 * i0 + 0], ay = sh_verts[3 * i0 + 1], az = sh_verts[3 * i0 + 2];
        float bx = sh_verts[3 * i1 + 0], by = sh_verts[3 * i1 + 1], bz = sh_verts[3 * i1 + 2];
        float cx = sh_verts[3 * i2 + 0], cy = sh_verts[3 * i2 + 1], cz = sh_verts[3 * i2 + 2];
        sh_pre[0][t] = ax;      sh_pre[1][t] = ay;      sh_pre[2][t] = az;
        sh_pre[3][t] = bx - ax; sh_pre[4][t] = by - ay; sh_pre[5][t] = bz - az;
        sh_pre[6][t] = cx - ax; sh_pre[7][t] = cy - ay; sh_pre[8][t] = cz - az;
    }
    __syncthreads();

    // ---- Stage 2: one wave per (tx, rx, candidate) ----
    const u32 lane  = threadIdx.x & 31u;
    const int combo = (int)(blockIdx.x * (blockDim.x >> 5) + (threadIdx.x >> 5));
    if (combo >= total) return;
    const int p   = combo % npc;
    const int rxi = (combo / npc) % nrx;
    const int txi = combo / (npc * nrx);

    const int t0 = g_pc[p * PATH_ORDER + 0];
    const int t1 = g_pc[p * PATH_ORDER + 1];

    const V3 txp = v3(g_tx[3 * txi + 0], g_tx[3 * txi + 1], g_tx[3 * txi + 2]);
    const V3 rxp = v3(g_rx[3 * rxi + 0], g_rx[3 * rxi + 1], g_rx[3 * rxi + 2]);
    const V3 n0  = v3(g_norms[3 * t0 + 0], g_norms[3 * t0 + 1], g_norms[3 * t0 + 2]);
    const V3 n1  = v3(g_norms[3 * t1 + 0], g_norms[3 * t1 + 1], g_norms[3 * t1 + 2]);

    const V3 a0  = v3(sh_pre[0][t0], sh_pre[1][t0], sh_pre[2][t0]);  // mirror vertex 0
    const V3 e10 = v3(sh_pre[3][t0], sh_pre[4][t0], sh_pre[5][t0]);
    const V3 e20 = v3(sh_pre[6][t0], sh_pre[7][t0], sh_pre[8][t0]);
    const V3 a1  = v3(sh_pre[0][t1], sh_pre[1][t1], sh_pre[2][t1]);  // mirror vertex 1
    const V3 e11 = v3(sh_pre[3][t1], sh_pre[4][t1], sh_pre[5][t1]);
    const V3 e21 = v3(sh_pre[6][t1], sh_pre[7][t1], sh_pre[8][t1]);

    // image method (order 2): forward images, backward plane intersections
    const V3 img1 = mirror_pt(txp, a0, n0);
    const V3 img2 = mirror_pt(img1, a1, n1);
    const V3 p2   = plane_line(rxp, img2, a1, n1);   // reflection point on tri1
    const V3 p1   = plane_line(p2,  img1, a0, n0);   // reflection point on tri0

    // mask_1: reflection points inside their triangles
    const bool m1 = tri_contains(a0, vadd(a0, e10), vadd(a0, e20), p1) &&
                    tri_contains(a1, vadd(a1, e11), vadd(a1, e21), p2);

    // mask_2: consecutive path points on same side of each mirror plane
    const float s0 = vdot(vsub(txp, a0), n0) * vdot(vsub(p2,  a0), n0);
    const float s1 = vdot(vsub(p1,  a1), n1) * vdot(vsub(rxp, a1), n1);
    const bool m2 = (s0 >= 0.0f) && (s1 >= 0.0f);

    // mask_3: no segment occluded (Moller-Trumbore any-hit, lanes split triangles)
    const V3 fp0 = txp, fp1 = p1, fp2 = p2, fp3 = rxp;
    bool anyHit = false;
#pragma unroll
    for (int k = 0; k < 3; ++k) {
        if (!anyHit) {                       // uniform early-out (ballot result is uniform)
            const V3 ro = (k == 0) ? fp0 : ((k == 1) ? fp1 : fp2);
            const V3 nx = (k == 0) ? fp1 : ((k == 1) ? fp2 : fp3);
            const V3 rd = vsub(nx, ro);
            bool local = false;
            for (u32 t = lane; t < N_TRI; t += 32u) {
                V3 tv0 = v3(sh_pre[0][t], sh_pre[1][t], sh_pre[2][t]);
                V3 te1 = v3(sh_pre[3][t], sh_pre[4][t], sh_pre[5][t]);
                V3 te2 = v3(sh_pre[6][t], sh_pre[7][t], sh_pre[8][t]);
                V3 h = vcross(rd, te2);
                float aDet = vdot(te1, h);
                float f = 1.0f / aDet;
                V3 s = vsub(ro, tv0);
                float u = f * vdot(s, h);
                V3 q = vcross(s, te1);
                float v = f * vdot(rd, q);
                float tt = f * vdot(te2, q);
                local |= (__builtin_fabsf(aDet) > MT_EPS) & (u >= 0.0f) & (u <= 1.0f) &
                         (v >= 0.0f) & ((u + v) <= 1.0f) & (tt > MT_EPS) & (tt < MT_HIT_T);
            }
#if __has_builtin(__builtin_amdgcn_ballot_w32)
            anyHit = anyHit || (__builtin_amdgcn_ballot_w32(local) != 0u);
#else
            anyHit = anyHit || (__ballot(local) != 0ull);
#endif
        }
    }

    if (lane == 0u) {
        const bool ok = m1 && m2 && !anyHit;
        float* fpOut = out + (size_t)combo * 12u;            // full_paths [.,4,3]
        fpOut[0]  = fp0.x; fpOut[1]  = fp0.y; fpOut[2]  = fp0.z;
        fpOut[3]  = fp1.x; fpOut[4]  = fp1.y; fpOut[5]  = fp1.z;
        fpOut[6]  = fp2.x; fpOut[7]  = fp2.y; fpOut[8]  = fp2.z;
        fpOut[9]  = fp3.x; fpOut[10] = fp3.y; fpOut[11] = fp3.z;
        float* objOut = out + (size_t)total * 12u + (size_t)combo * 4u;  // objects [.,4]
        objOut[0] = (float)txi; objOut[1] = (float)t0;
        objOut[2] = (float)t1;  objOut[3] = (float)rxi;
        out[(size_t)total * 16u + (size_t)combo] = ok ? 1.0f : 0.0f;     // mask
    }
}

// ---------------------------------------------------------------------------
extern "C" void kernel_launch(void* const* d_in, const int* in_sizes, int n_in,
                              void* d_out, int out_size, void* d_ws, size_t ws_size,
                              hipStream_t stream) {
    (void)n_in; (void)out_size; (void)d_ws; (void)ws_size;
    const float* g_tx    = (const float*)d_in[0];  // transmitters [T,3]
    const float* g_rx    = (const float*)d_in[1];  // receivers    [R,3]
    const float* g_verts = (const float*)d_in[2];  // vertices     [900,3]
    const float* g_norms = (const float*)d_in[3];  // normals      [400,3]
    const int*   g_tris  = (const int*)d_in[4];    // triangles    [400,3]
    const int*   g_pc    = (const int*)d_in[5];    // path_candidates [2048,2]

    const int ntx   = in_sizes[0] / 3;
    const int nrx   = in_sizes[1] / 3;
    const int npc   = in_sizes[5] / PATH_ORDER;
    const int total = ntx * nrx * npc;             // 8192 combos

    const int block = 256;                          // 8 wave32s
    const int wavesPerBlock = block / 32;
    const int grid = (total + wavesPerBlock - 1) / wavesPerBlock;   // 1024 blocks

    rt_paths_kernel<<<grid, block, 0, stream>>>(g_tx, g_rx, g_verts, g_norms,
                                                g_tris, g_pc, (float*)d_out,
                                                ntx, nrx, npc, total);
}